// GraphFeatureEncoder_31327491457576
// MI455X (gfx1250) — compile-verified
//
#include <hip/hip_runtime.h>

typedef float v2f __attribute__((ext_vector_type(2)));
typedef float v8f __attribute__((ext_vector_type(8)));

#define NNODES 30000
#define NEDGES 480000
#define CH 64
#define NH 4
#define BN_EPS 1e-5f

// ---------------------------------------------------------------- utilities
__global__ void fill_kernel(float* __restrict__ p, float v, int n) {
  int i = blockIdx.x * blockDim.x + threadIdx.x;
  if (i < n) p[i] = v;
}

__global__ void count_kernel(const int* __restrict__ dst, float* __restrict__ cnt, int nE) {
  int i = blockIdx.x * blockDim.x + threadIdx.x;
  if (i < nE) atomicAdd(&cnt[dst[i]], 1.0f);
}

// ---------------------------------------------------------------- edge kernel
// One wave32 handles a tile of 16 (real or virtual self-loop) edges.
// P[16,256] = Xj[16,64] @ W.T  via v_wmma_f32_16x16x4_f32 (fp32, K=4),
// then attention-weighted head combine and atomic scatter into agg[dst].
__launch_bounds__(128)
__global__ void feast_edge_kernel(const float* __restrict__ x,
                                  const int* __restrict__ src,
                                  const int* __restrict__ dst,
                                  const float* __restrict__ W,   // [256,64] row-major
                                  const float* __restrict__ U,   // [4,64]
                                  const float* __restrict__ cv,  // [4]
                                  float* __restrict__ agg,       // [N,64]
                                  int nE, int nTiles)
{
  // Pre-swizzled B fragments: sWB[nt][k][lane] = (W[col][kk], W[col][kk+1])
  // col = nt*16 + (lane&15), kk = 4k + 2*(lane>>4).  16*16*32 float2 = 64KB.
  __shared__ float2 sWB[16 * 16 * 32];
  const int tid = threadIdx.x;
  for (int idx = tid; idx < 16 * 16 * 32; idx += blockDim.x) {
    int lane = idx & 31;
    int k    = (idx >> 5) & 15;
    int nt   = idx >> 9;
    int col  = nt * 16 + (lane & 15);
    int kk   = 4 * k + 2 * (lane >> 4);
    sWB[idx] = make_float2(W[col * CH + kk], W[col * CH + kk + 1]);
  }
  __syncthreads();

  const int wave = tid >> 5;
  const int lane = tid & 31;
  const int tile = blockIdx.x * (blockDim.x >> 5) + wave;
  if (tile >= nTiles) return;

  const int row = lane & 15;       // edge within tile (M index)
  const int hi  = lane >> 4;       // K-half selector
  const int e   = tile * 16 + row;

  int s, d;
  if (e < nE) { s = src[e]; d = dst[e]; }
  else        { s = e - nE; d = s; }          // virtual self-loop edge

  const float* __restrict__ xj = x + (long)s * CH;
  const float* __restrict__ xi = x + (long)d * CH;

  // ---- gather A fragments + attention logits (lane pair l / l+16 per edge)
  v2f   aj[16];
  float lg[NH] = {0.f, 0.f, 0.f, 0.f};
  const int choff = 2 * hi;
#pragma unroll
  for (int k = 0; k < 16; ++k) {
    int ch = 4 * k + choff;
    float2 vj = *(const float2*)(xj + ch);
    float2 vi = *(const float2*)(xi + ch);
    aj[k].x = vj.x; aj[k].y = vj.y;
    float dx0 = vi.x - vj.x;
    float dx1 = vi.y - vj.y;
#pragma unroll
    for (int m = 0; m < NH; ++m) {
      float2 u = *(const float2*)(U + m * CH + ch);
      lg[m] += dx0 * u.x + dx1 * u.y;
    }
  }
  // reduce partial dots across the lane pair, add c, softmax over heads
  float attn[NH];
  float mx = -1e30f;
#pragma unroll
  for (int m = 0; m < NH; ++m) {
    lg[m] += __shfl_xor(lg[m], 16, 32);
    lg[m] += cv[m];
    mx = fmaxf(mx, lg[m]);
  }
  float ssum = 0.f;
#pragma unroll
  for (int m = 0; m < NH; ++m) { attn[m] = __expf(lg[m] - mx); ssum += attn[m]; }
  float rs = 1.0f / ssum;
#pragma unroll
  for (int m = 0; m < NH; ++m) attn[m] *= rs;

  // ---- GEMM + head-weighted accumulation: msg[16,64]
  v8f msgacc[4];
#pragma unroll
  for (int c4 = 0; c4 < 4; ++c4)
#pragma unroll
    for (int j = 0; j < 8; ++j) msgacc[c4][j] = 0.f;

#pragma unroll
  for (int m = 0; m < NH; ++m) {
    // attention of edge (j + 8*hi), head m (C-layout row mapping)
    float am[8];
#pragma unroll
    for (int j = 0; j < 8; ++j) am[j] = __shfl(attn[m], j + 8 * hi, 32);

#pragma unroll
    for (int c4 = 0; c4 < 4; ++c4) {
      const int nt = m * 4 + c4;
      v8f acc;
#pragma unroll
      for (int j = 0; j < 8; ++j) acc[j] = 0.f;
      const float2* bp = &sWB[nt * 512 + lane];
#pragma unroll
      for (int k = 0; k < 16; ++k) {
        float2 bf = bp[k * 32];
        v2f B; B.x = bf.x; B.y = bf.y;
        acc = __builtin_amdgcn_wmma_f32_16x16x4_f32(
            false, aj[k], false, B, (short)0, acc, false, false);
      }
#pragma unroll
      for (int j = 0; j < 8; ++j) msgacc[c4][j] += am[j] * acc[j];
    }
  }

  // ---- scatter: agg[dst[e]][o] += msg[e][o]
  const int col0 = lane & 15;
#pragma unroll
  for (int j = 0; j < 8; ++j) {
    int dj = __shfl(d, j + 8 * hi, 32);
    float* base = agg + (long)dj * CH + col0;
    atomicAdd(base +  0, msgacc[0][j]);
    atomicAdd(base + 16, msgacc[1][j]);
    atomicAdd(base + 32, msgacc[2][j]);
    atomicAdd(base + 48, msgacc[3][j]);
  }
}

// ------------------------------------------- mean + bias + relu + BN stats
__global__ void finalize_bn_stats_kernel(const float* __restrict__ agg,
                                         const float* __restrict__ cnt,
                                         const float* __restrict__ b,
                                         float* __restrict__ hpre,
                                         float* __restrict__ colsum,
                                         float* __restrict__ colsq,
                                         int nNodes)
{
  const int o = threadIdx.x & 63;
  const int r = threadIdx.x >> 6;   // 0..3 (blockDim = 256)
  const float bo = b[o];
  float acc = 0.f, acc2 = 0.f;
  for (int n = blockIdx.x * 4 + r; n < nNodes; n += gridDim.x * 4) {
    float v = agg[(long)n * CH + o] / cnt[n] + bo;
    v = fmaxf(v, 0.f);
    hpre[(long)n * CH + o] = v;
    acc += v; acc2 += v * v;
  }
  __shared__ float s1[4][64];
  __shared__ float s2[4][64];
  s1[r][o] = acc; s2[r][o] = acc2;
  __syncthreads();
  if (r == 0) {
    float t1 = s1[0][o] + s1[1][o] + s1[2][o] + s1[3][o];
    float t2 = s2[0][o] + s2[1][o] + s2[2][o] + s2[3][o];
    atomicAdd(&colsum[o], t1);
    atomicAdd(&colsq[o], t2);
  }
}

__global__ void bn_kernel(float* __restrict__ h,
                          const float* __restrict__ colsum,
                          const float* __restrict__ colsq,
                          const float* __restrict__ g,
                          const float* __restrict__ bt,
                          int nNodes)
{
  int i = blockIdx.x * blockDim.x + threadIdx.x;
  if (i >= nNodes * CH) return;
  int o = i & 63;
  float rn  = 1.0f / (float)nNodes;
  float mu  = colsum[o] * rn;
  float var = colsq[o] * rn - mu * mu;   // biased variance (training-mode BN)
  float inv = rsqrtf(var + BN_EPS);
  h[i] = (h[i] - mu) * inv * g[o] + bt[o];
}

__global__ void finalize_last_kernel(const float* __restrict__ agg,
                                     const float* __restrict__ cnt,
                                     const float* __restrict__ b,
                                     float* __restrict__ out,
                                     int nNodes)
{
  int i = blockIdx.x * blockDim.x + threadIdx.x;
  if (i >= nNodes * CH) return;
  out[i] = agg[i] / cnt[i >> 6] + b[i & 63];
}

// ---------------------------------------------------------------- launcher
extern "C" void kernel_launch(void* const* d_in, const int* in_sizes, int n_in,
                              void* d_out, int out_size, void* d_ws, size_t ws_size,
                              hipStream_t stream)
{
  (void)in_sizes; (void)n_in; (void)out_size; (void)ws_size;

  const float* x   = (const float*)d_in[0];
  const int*   ei  = (const int*)d_in[1];
  const int*   src = ei;
  const int*   dst = ei + NEDGES;
  const float* W[3]  = {(const float*)d_in[2],  (const float*)d_in[8],  (const float*)d_in[14]};
  const float* U[3]  = {(const float*)d_in[3],  (const float*)d_in[9],  (const float*)d_in[15]};
  const float* c[3]  = {(const float*)d_in[4],  (const float*)d_in[10], (const float*)d_in[16]};
  const float* b[3]  = {(const float*)d_in[5],  (const float*)d_in[11], (const float*)d_in[17]};
  const float* g[2]  = {(const float*)d_in[6],  (const float*)d_in[12]};
  const float* bt[2] = {(const float*)d_in[7],  (const float*)d_in[13]};

  // workspace layout (floats): agg[N*64] | colsum[64] | colsq[64] | cnt[N] | hA[N*64] | hB[N*64]
  float* ws     = (float*)d_ws;
  float* agg    = ws;
  float* colsum = agg + (size_t)NNODES * CH;
  float* cnt    = colsum + 128;
  float* hA     = cnt + NNODES;
  float* hB     = hA + (size_t)NNODES * CH;

  const int nTot = NNODES * CH;

  // in-degree counts (self-loop baked in via init=1)
  fill_kernel<<<(NNODES + 255) / 256, 256, 0, stream>>>(cnt, 1.0f, NNODES);
  count_kernel<<<(NEDGES + 255) / 256, 256, 0, stream>>>(dst, cnt, NEDGES);

  const int nTiles = (NEDGES + NNODES) / 16;            // 31875, exact
  const int wavesPerBlock = 4;                          // blockDim 128 = 4 waves
  const int edgeBlocks = (nTiles + wavesPerBlock - 1) / wavesPerBlock;

  const float* hin = x;
  float* houts[2] = {hA, hB};
  for (int l = 0; l < 3; ++l) {
    // zero agg + colsum + colsq (contiguous)
    fill_kernel<<<(nTot + 128 + 255) / 256, 256, 0, stream>>>(agg, 0.0f, nTot + 128);

    feast_edge_kernel<<<edgeBlocks, 128, 0, stream>>>(
        hin, src, dst, W[l], U[l], c[l], agg, NEDGES, nTiles);

    if (l < 2) {
      finalize_bn_stats_kernel<<<256, 256, 0, stream>>>(
          agg, cnt, b[l], houts[l], colsum, colsum + 64, NNODES);
      bn_kernel<<<(nTot + 255) / 256, 256, 0, stream>>>(
          houts[l], colsum, colsum + 64, g[l], bt[l], NNODES);
      hin = houts[l];
    } else {
      finalize_last_kernel<<<(nTot + 255) / 256, 256, 0, stream>>>(
          agg, cnt, b[2], (float*)d_out, NNODES);
    }
  }
}